// PositionalEncoderQuantum_65481071400113
// MI455X (gfx1250) — compile-verified
//
#include <hip/hip_runtime.h>
#include <stdint.h>

typedef float v2f __attribute__((ext_vector_type(2)));
typedef float v4f __attribute__((ext_vector_type(4)));
typedef float v8f __attribute__((ext_vector_type(8)));

// ---- gfx1250 async global->LDS copy (ASYNCcnt-tracked) ------------------
#define ASYNC_WAIT(n) asm volatile("s_wait_asynccnt " #n ::: "memory")

__device__ __forceinline__ void async_b128(uint32_t lds_off, uint32_t voff_bytes,
                                           const void* sbase) {
  // vdst = LDS byte address VGPR, vaddr = 32-bit byte offset, saddr = 64-bit base
  asm volatile("global_load_async_to_lds_b128 %0, %1, %2"
               :: "v"(lds_off), "v"(voff_bytes), "s"(sbase)
               : "memory");
}

// =========================================================================
// Kernel 1: theta reduction via WMMA f32 16x16x4 chains (B = ones).
//   angles: [32768, 1024] f32 ; phase(out, ws): [32768, 8] f32 = cos(theta)
// One wave per 16-row tile. 32 chunks of 32 cols; chunk t contributes one
// K=4 WMMA per wire w (terms = columns {w, w+8, w+16, w+24} of the chunk).
// B==1 everywhere -> result invariant to hardware K-slot permutation.
// Steady-state software pipeline (3 LDS buffers, depth-2 prefetch), loop
// deliberately NOT unrolled to keep RA pressure ~110 VGPRs (no spills).
// =========================================================================
#define K1_WAVES   8
#define K1_STRIDE  36                   // floats per staged row (144B, 16B-aligned)
#define K1_BUFSZ   (16 * K1_STRIDE)     // 576 floats per chunk buffer
#define K1_BUFBYTES (K1_BUFSZ * 4)
#define K1_NBUF    3

__global__ __launch_bounds__(256) void theta_wmma_kernel(
    const float* __restrict__ angles, float* __restrict__ phase) {
  __shared__ __align__(16) float stage[K1_WAVES * K1_NBUF * K1_BUFSZ]; // 54 KB

  const int lane = threadIdx.x & 31;
  const int wave = threadIdx.x >> 5;
  const int tile = blockIdx.x * K1_WAVES + wave;   // 0..2047
  const int row0 = tile * 16;
  const int r    = lane & 15;
  const int h    = lane >> 4;

  // Per-lane async addressing: issue q covers rows q*4..q*4+3, 128B per row.
  const int arow = lane >> 3;       // 0..3
  const int acol = (lane & 7) * 4;  // float col within 32-col chunk
  const uint32_t ldsWave =
      (uint32_t)(size_t)(&stage[0]) + (uint32_t)(wave * K1_NBUF * K1_BUFSZ) * 4u;

  // Running global byte offsets (advance by 128B per chunk) and fixed
  // per-issue LDS offsets (buffer index added separately).
  uint32_t voffQ[4], ldsQ[4];
#pragma unroll
  for (int q = 0; q < 4; ++q) {
    const int rowRel = q * 4 + arow;
    voffQ[q] = (uint32_t)(((row0 + rowRel) * 1024 + acol) * 4);
    ldsQ[q]  = ldsWave + (uint32_t)((rowRel * K1_STRIDE + acol) * 4);
  }

  // Prologue: prefetch chunks 0 (buf 0) and 1 (buf 1).
#pragma unroll
  for (int c = 0; c < 2; ++c) {
#pragma unroll
    for (int q = 0; q < 4; ++q)
      async_b128(ldsQ[q] + (uint32_t)(c * K1_BUFBYTES),
                 voffQ[q] + (uint32_t)(c * 128), angles);
  }

  v8f acc[8];
#pragma unroll
  for (int w = 0; w < 8; ++w) acc[w] = (v8f){0, 0, 0, 0, 0, 0, 0, 0};

  v2f bones; bones.x = 1.0f; bones.y = 1.0f;  // K-permutation-proof B matrix

  const float* stBase = &stage[wave * K1_NBUF * K1_BUFSZ + r * K1_STRIDE + h * 16];

  uint32_t voffCur = 2u * 128u;  // next chunk to issue = chunk 2
  int issueBuf = 2;              // (chunk 2) % 3
  int procBuf  = 0;

#pragma unroll 1
  for (int t = 0; t < 30; ++t) {
    // Issue chunk t+2 into issueBuf.
#pragma unroll
    for (int q = 0; q < 4; ++q)
      async_b128(ldsQ[q] + (uint32_t)(issueBuf * K1_BUFBYTES),
                 voffQ[q] + voffCur, angles);
    voffCur += 128u;
    issueBuf = (issueBuf == 2) ? 0 : issueBuf + 1;

    ASYNC_WAIT(8);  // 3 chunks in flight -> oldest (t) complete

    const float* st = stBase + procBuf * K1_BUFSZ;
#pragma unroll
    for (int w = 0; w < 8; ++w) {
      v2f a;
      a.x = st[w];
      a.y = st[w + 8];
      acc[w] = __builtin_amdgcn_wmma_f32_16x16x4_f32(
          false, a, false, bones, (short)0, acc[w], false, false);
    }
    procBuf = (procBuf == 2) ? 0 : procBuf + 1;
  }

  // Peeled tail: chunks 30 and 31 (no further issues).
#pragma unroll 1
  for (int tail = 0; tail < 2; ++tail) {
    if (tail == 0) { ASYNC_WAIT(4); } else { ASYNC_WAIT(0); }
    const float* st = stBase + procBuf * K1_BUFSZ;
#pragma unroll
    for (int w = 0; w < 8; ++w) {
      v2f a;
      a.x = st[w];
      a.y = st[w + 8];
      acc[w] = __builtin_amdgcn_wmma_f32_16x16x4_f32(
          false, a, false, bones, (short)0, acc[w], false, false);
    }
    procBuf = (procBuf == 2) ? 0 : procBuf + 1;
  }

  // D layout: element j of acc = row (j + 8*h); every N-column identical.
  if (r == 0) {
    const int m0 = h * 8;
#pragma unroll
    for (int w = 0; w < 8; ++w) {
#pragma unroll
      for (int j = 0; j < 8; ++j)
        phase[(row0 + m0 + j) * 8 + w] = cosf(acc[w][j]);
    }
  }
}

// =========================================================================
// Kernel 2: out = x + pe(s,d) + cos_theta[row, d%8], streaming f32.
// Block = 16 s-values x 8 batches x 256-col slice. Thread owns one fixed
// float4 column -> pe sincos computed once per (s, colpair), reused over
// all 8 batches. Non-temporal x/out (read/write-once streams).
// =========================================================================
__global__ __launch_bounds__(256) void add_pe_phase_kernel(
    const float* __restrict__ x, const float* __restrict__ phase,
    float* __restrict__ out) {
  __shared__ __align__(16) float ph[128 * 8];  // cos-theta for this block's rows

  const int t      = threadIdx.x;
  const int sBlock = blockIdx.x >> 2;  // 0..255
  const int slice  = blockIdx.x & 3;   // 256-col slice
  const int s0     = sBlock * 16;

  {  // preload phase rows: (b, sLocal, half) <- thread, fully coalesced
    const int b    = t >> 5;
    const int sl   = (t >> 1) & 15;
    const int half = t & 1;
    const v4f v = *(const v4f*)&phase[((size_t)(b * 4096 + s0 + sl)) * 8 + half * 4];
    *(v4f*)&ph[(b * 16 + sl) * 8 + half * 4] = v;
  }
  __syncthreads();

  const int cg   = t & 63;
  const int ssub = t >> 6;
  const int c4   = slice * 256 + cg * 4;  // fixed float4 column for this thread

  const float PEK = -9.210340371976184e+00f / 1024.0f;  // -ln(10000)/D
  const float ea = expf(PEK * (float)c4);
  const float eb = expf(PEK * (float)(c4 + 2));

#pragma unroll
  for (int i = 0; i < 4; ++i) {
    const int sl = ssub * 4 + i;
    const int s  = s0 + sl;
    float sa, ca, sb, cb;
    sincosf((float)s * ea, &sa, &ca);
    sincosf((float)s * eb, &sb, &cb);
    const v4f pe = {sa, ca, sb, cb};  // cols c4(even)->sin, c4+1->cos, ...
#pragma unroll 4
    for (int b = 0; b < 8; ++b) {
      const size_t idx = (size_t)(b * 4096 + s) * 1024 + (size_t)c4;
      v4f xv = __builtin_nontemporal_load((const v4f*)&x[idx]);
      const v4f pv = *(const v4f*)&ph[(b * 16 + sl) * 8 + (c4 & 4)];
      v4f o = xv + pe + pv;
      __builtin_nontemporal_store(o, (v4f*)&out[idx]);
    }
  }
}

// =========================================================================
extern "C" void kernel_launch(void* const* d_in, const int* in_sizes, int n_in,
                              void* d_out, int out_size, void* d_ws, size_t ws_size,
                              hipStream_t stream) {
  (void)in_sizes; (void)n_in; (void)out_size; (void)ws_size;
  const float* x      = (const float*)d_in[0];  // [8,4096,1024] f32
  const float* angles = (const float*)d_in[1];  // [32768,1024] f32
  float* out   = (float*)d_out;                 // [8,4096,1024] f32
  float* phase = (float*)d_ws;                  // [32768,8] f32 scratch (1 MiB)

  // 2048 row-tiles of 16 rows, one per wave, 8 waves/block.
  theta_wmma_kernel<<<256, 256, 0, stream>>>(angles, phase);
  // 256 s-blocks x 4 col-slices.
  add_pe_phase_kernel<<<1024, 256, 0, stream>>>(x, phase, out);
}